// EmbedMLSADF_SVS_28398323761316
// MI455X (gfx1250) — compile-verified
//
#include <hip/hip_runtime.h>
#include <math.h>

typedef float v2f __attribute__((ext_vector_type(2)));
typedef float v8f __attribute__((ext_vector_type(8)));

namespace cfg {
constexpr int B = 4, T = 200, HOP = 120, N = T * HOP;   // N = 24000
constexpr int M1 = 40, A1 = 5, HID = 256, COND = 192;
constexpr int BT = B * T;                               // 800 rows
constexpr int KF = 48;                                  // padded feature K (46 -> 48)
constexpr int NM = 48;                                  // padded mcep cols (40 -> 48)
constexpr float SR = 24000.f, ALPHA = 0.466f;
constexpr float AA = 1.f - ALPHA * ALPHA;
constexpr float F0_MEAN = 5.f, F0_STD = 0.25f;
constexpr float MCEP_MEAN = 0.f, MCEP_STD = 1.f;
constexpr float APDC_MEAN = 0.f, APDC_STD = 1.f;
constexpr float NOISE_RATIO = 0.5f, SP_RATE = 1.f, F0_RATE = 1.f;
}

// ---------------------------------------------------------------------------
// Packing kernels (zero-pad K/N dims to multiples of 4/16 for WMMA)
// ---------------------------------------------------------------------------
__global__ void pack_feat(const float* __restrict__ mceps,
                          const float* __restrict__ apdcs,
                          const float* __restrict__ f0s,
                          float* __restrict__ featP) {
  int id = blockIdx.x * blockDim.x + threadIdx.x;
  if (id >= cfg::BT * cfg::KF) return;
  int row = id / cfg::KF, c = id % cfg::KF;
  float v = 0.f;
  if (c < 40)      v = mceps[row * cfg::M1 + c];
  else if (c < 45) v = apdcs[row * cfg::A1 + (c - 40)];
  else if (c == 45) v = f0s[row];
  featP[id] = v;
}

__global__ void pack_w1(const float* __restrict__ Wm, const float* __restrict__ bm,
                        const float* __restrict__ Wa, const float* __restrict__ ba,
                        const float* __restrict__ Wf, const float* __restrict__ bf,
                        float* __restrict__ WcatP, float* __restrict__ biasH) {
  int id = blockIdx.x * blockDim.x + threadIdx.x;
  if (id >= cfg::KF * cfg::HID) return;
  int c = id / cfg::HID, n = id % cfg::HID;
  float v = 0.f;
  if (c < 40)      v = Wm[c * cfg::HID + n];
  else if (c < 45) v = Wa[(c - 40) * cfg::HID + n];
  else if (c == 45) v = Wf[n];
  WcatP[id] = v;
  if (id < cfg::HID) biasH[id] = bm[id] + ba[id] + bf[id];
}

__global__ void pack_w3(const float* __restrict__ Wsrc, const float* __restrict__ bsrc,
                        float* __restrict__ Wdst, float* __restrict__ bdst) {
  int id = blockIdx.x * blockDim.x + threadIdx.x;
  if (id >= cfg::COND * cfg::NM) return;
  int k = id / cfg::NM, n = id % cfg::NM;
  Wdst[id] = (n < cfg::M1) ? Wsrc[k * cfg::M1 + n] : 0.f;
  if (id < cfg::NM) bdst[id] = (id < cfg::M1) ? bsrc[id] : 0.f;
}

__global__ void apw_kernel(const float* __restrict__ apdcs,
                           const float* __restrict__ Wap,
                           float* __restrict__ apW) {
  int id = blockIdx.x * blockDim.x + threadIdx.x;
  if (id >= cfg::BT * cfg::NM) return;
  int row = id / cfg::NM, c = id % cfg::NM;
  float v = 0.f;
  if (c < cfg::M1) {
#pragma unroll
    for (int j = 0; j < cfg::A1; ++j)
      v = fmaf(apdcs[row * cfg::A1 + j] * cfg::APDC_STD + cfg::APDC_MEAN,
               Wap[j * cfg::M1 + c], v);
  }
  apW[id] = v;
}

// ---------------------------------------------------------------------------
// WMMA fp32 GEMM: one wave per 16x16 output tile, K stepped by 4.
// mode 0: C = acc + bias[col]
// mode 1: C = 0.1*tanh(acc+bias) + sp_rate*sp[row,col<40] (+ add2[row,col])
// ---------------------------------------------------------------------------
__global__ __launch_bounds__(32)
void wmma_gemm16(const float* __restrict__ A, int lda,
                 const float* __restrict__ Bm, int ldb,
                 const float* __restrict__ bias,
                 float* __restrict__ C, int ldc, int K, int mode,
                 const float* __restrict__ sp, const float* __restrict__ add2) {
  const int lane = threadIdx.x & 31;
  const int half = lane >> 4;      // 0: K pair {0,1}, 1: K pair {2,3}
  const int l15  = lane & 15;      // M (for A) / N (for B,C)
  const int tm = blockIdx.x * 16;
  const int tn = blockIdx.y * 16;

  v8f acc = {0.f, 0.f, 0.f, 0.f, 0.f, 0.f, 0.f, 0.f};
  const float* arow = A + (tm + l15) * lda;
  for (int k = 0; k < K; k += 4) {
    const int ka = k + half * 2;
    v2f a, b;
    a[0] = arow[ka];
    a[1] = arow[ka + 1];
    b[0] = Bm[ka * ldb + tn + l15];
    b[1] = Bm[(ka + 1) * ldb + tn + l15];
    acc = __builtin_amdgcn_wmma_f32_16x16x4_f32(false, a, false, b,
                                                (short)0, acc, false, false);
  }

  const int col = tn + l15;
  const float bv = bias ? bias[col] : 0.f;
#pragma unroll
  for (int r = 0; r < 8; ++r) {
    const int row = tm + r + half * 8;          // C layout: v[r] = rows r / r+8
    float v = acc[r] + bv;
    if (mode == 1) {
      float t = 0.1f * tanhf(v);
      float spv = (col < cfg::M1)
          ? (sp[row * cfg::M1 + col] * cfg::MCEP_STD + cfg::MCEP_MEAN) * cfg::SP_RATE
          : 0.f;
      float a2 = add2 ? add2[row * cfg::NM + col] : 0.f;
      v = t + spv + a2;
    }
    C[row * ldc + col] = v;
  }
}

// ---------------------------------------------------------------------------
// mc2b: b[m] = mc[m] - alpha*b[m+1], top-down. One thread per (path,row).
// Output layout bcoef[channel=path*4+batch][t][M1]  (matches MLSA kernel).
// ---------------------------------------------------------------------------
__global__ void mc2b_kernel(const float* __restrict__ mcp,
                            const float* __restrict__ mca,
                            float* __restrict__ bcoef) {
  int id = blockIdx.x * blockDim.x + threadIdx.x;
  if (id >= 2 * cfg::BT) return;
  int path = id / cfg::BT, row = id % cfg::BT;
  const float* mc = (path ? mca : mcp) + row * cfg::NM;
  float bb[cfg::M1];
  float carry = 0.f;
#pragma unroll
  for (int m = cfg::M1 - 1; m >= 0; --m) {
    bb[m] = mc[m] - cfg::ALPHA * carry;
    carry = bb[m];
  }
  int batch = row / cfg::T, t = row % cfg::T;
  float* dst = bcoef + ((path * cfg::B + batch) * cfg::T + t) * cfg::M1;
#pragma unroll
  for (int m = 0; m < cfg::M1; ++m) dst[m] = bb[m];
}

// ---------------------------------------------------------------------------
// Per-batch frame-level phase prefix sum (200 steps, 4 chains).
// ---------------------------------------------------------------------------
__global__ __launch_bounds__(32)
void f0base_kernel(const float* __restrict__ f0s,
                   float* __restrict__ f0buf, float* __restrict__ base) {
  int b = threadIdx.x;
  if (b >= cfg::B) return;
  float acc = 0.f;
  for (int t = 0; t < cfg::T; ++t) {
    float f0 = expf(f0s[b * cfg::T + t] * cfg::F0_STD + cfg::F0_MEAN) * cfg::F0_RATE;
    f0buf[b * cfg::T + t] = f0;
    base[b * cfg::T + t] = acc;             // phase at end of previous frame
    acc += (float)cfg::HOP * f0 / cfg::SR;
  }
}

// Excitation: phase is affine within a hop -> fully parallel pulse detection.
__global__ void exc_kernel(const float* __restrict__ f0buf,
                           const float* __restrict__ base,
                           const float* __restrict__ noise,
                           float* __restrict__ exc) {
  int id = blockIdx.x * blockDim.x + threadIdx.x;
  if (id >= cfg::B * cfg::N) return;
  int b = id / cfg::N, n = id % cfg::N;
  int t = n / cfg::HOP, k = n % cfg::HOP;
  float f0 = f0buf[b * cfg::T + t];
  float bs = base[b * cfg::T + t];
  float inv = f0 / cfg::SR;
  float ph  = bs + (float)(k + 1) * inv;
  float php = (k > 0) ? bs + (float)k * inv : bs;   // last sample of prev frame
  float w = floorf(ph), wp = floorf(php);
  float amp = sqrtf(cfg::SR / fmaxf(f0, 1.f));
  float pulse = (w > wp) ? amp : 0.f;
  exc[b * cfg::N + n] = pulse * (1.f - cfg::NOISE_RATIO);               // periodic
  exc[(cfg::B + b) * cfg::N + n] = noise[b * cfg::N + n] * cfg::NOISE_RATIO; // aper
}

// ---------------------------------------------------------------------------
// MLSA synthesis scan: ONE wave32. lane = channel(8) * 4 + pade_tap(4).
// Order recurrence (38 deep) lives in registers; Pade reductions are
// shfl_xor within groups of 4; periodic+aperiodic sum via shfl_xor(16).
// ---------------------------------------------------------------------------
__global__ __launch_bounds__(32)
void mlsa_kernel(const float* __restrict__ exc,    // [8][N]
                 const float* __restrict__ bcoef,  // [8][T][M1]
                 float* __restrict__ out) {        // [B][N]
  const int lane = threadIdx.x;
  const int c = lane >> 2;          // channel 0..7 (0-3 periodic, 4-7 aperiodic)
  const int p = lane & 3;           // pade tap 0..3 (== reference taps 1..4)
  const int batch = c & 3;

  const float pade_tab[4] = {0.4999273f, 0.1067005f, 0.00956526f, 0.0003041358f};
  const float padep = pade_tab[p];
  const float signp = (p & 1) ? -1.f : 1.f;

  float d2[cfg::M1 + 1];
#pragma unroll
  for (int i = 0; i <= cfg::M1; ++i) d2[i] = 0.f;
  float bb[cfg::M1];
#pragma unroll
  for (int i = 0; i < cfg::M1; ++i) bb[i] = 0.f;

  float d1a_own = 0.f;   // d1a[p+1] (own tap's 'na' from previous step)
  float d1b_p   = 0.f;   // d1b[p]  = (p==0 ? x1_prev : npt_{p-1})
  float xp      = 0.f;   // pt2[p]  = (p==0 ? x2_prev : fy_{p-1})
  float eg = 1.f;

  const float* excC = exc + c * cfg::N;
  const float* bC   = bcoef + c * cfg::T * cfg::M1;

  for (int n = 0; n < cfg::N; ++n) {
    if ((n % cfg::HOP) == 0) {
      const int t = n / cfg::HOP;
      const float* bf = bC + t * cfg::M1;
#pragma unroll
      for (int m = 0; m < cfg::M1; ++m) bb[m] = bf[m];
      eg = expf(bb[0]);
      if (t + 1 < cfg::T)                       // warm L2/L0 for next frame
        __builtin_prefetch(bC + (t + 1) * cfg::M1, 0, 1);
    }

    float x = excC[n] * eg;

    // ---- stage 1 (mlsadf1, b[1] only) ----
    float na  = cfg::AA * d1b_p + cfg::ALPHA * d1a_own;
    float npt = na * bb[1];
    float v1  = npt * padep;
    float sv = v1 * signp;
    sv += __shfl_xor(sv, 1, 32); sv += __shfl_xor(sv, 2, 32);
    float sa = v1;
    sa += __shfl_xor(sa, 1, 32); sa += __shfl_xor(sa, 2, 32);
    float x1 = x + sv;
    float y1 = sa + x1;
    float npt_prev = __shfl(npt, lane - 1, 32);   // tap shift (unused at p==0)
    d1b_p = (p == 0) ? x1 : npt_prev;
    d1a_own = na;

    // ---- stage 2 (mlsafir over orders 2..M1, in-register, in-place shift) ----
    float d1 = cfg::AA * xp + cfg::ALPHA * d2[1];
    float prev = d1, dkeep = d2[2], y = 0.f;
#pragma unroll
    for (int m = 0; m < cfg::M1 - 2; ++m) {       // 38 iterations
      float dnext = d2[3 + m];
      float nw = dkeep + cfg::ALPHA * (dnext - prev);
      y = fmaf(nw, bb[2 + m], y);
      d2[3 + m] = nw;                             // new_d[3+m] = dupd[m]
      dkeep = dnext;
      prev = nw;
    }
    d2[0] = xp; d2[1] = d1; d2[2] = d1;

    float fy = y;
    float v2 = fy * padep;
    float sv2 = v2 * signp;
    sv2 += __shfl_xor(sv2, 1, 32); sv2 += __shfl_xor(sv2, 2, 32);
    float sa2 = v2;
    sa2 += __shfl_xor(sa2, 1, 32); sa2 += __shfl_xor(sa2, 2, 32);
    float x2 = y1 + sv2;
    float y2 = sa2 + x2;
    float fy_prev = __shfl(fy, lane - 1, 32);
    xp = (p == 0) ? x2 : fy_prev;

    float part = __shfl_xor(y2, 16, 32);          // partner channel (p<->a)
    if (lane < 16 && p == 0) out[batch * cfg::N + n] = y2 + part;
  }
}

// ---------------------------------------------------------------------------
// Host side
// ---------------------------------------------------------------------------
static inline int cdiv(int a, int b) { return (a + b - 1) / b; }

extern "C" void kernel_launch(void* const* d_in, const int* in_sizes, int n_in,
                              void* d_out, int out_size, void* d_ws, size_t ws_size,
                              hipStream_t stream) {
  using namespace cfg;
  const float* mceps = (const float*)d_in[0];
  const float* apdcs = (const float*)d_in[1];
  const float* f0s   = (const float*)d_in[2];
  const float* noise = (const float*)d_in[3];
  const float* Wm  = (const float*)d_in[4];  const float* bm  = (const float*)d_in[5];
  const float* Wa  = (const float*)d_in[6];  const float* ba  = (const float*)d_in[7];
  const float* Wf  = (const float*)d_in[8];  const float* bf  = (const float*)d_in[9];
  const float* Wpa = (const float*)d_in[10]; const float* bpa = (const float*)d_in[11];
  const float* Wpp = (const float*)d_in[12]; const float* bpp = (const float*)d_in[13];
  const float* Wrp = (const float*)d_in[14]; const float* brp = (const float*)d_in[15];
  const float* Wra = (const float*)d_in[16]; const float* bra = (const float*)d_in[17];
  const float* Wap = (const float*)d_in[18];
  float* out = (float*)d_out;

  float* ws = (float*)d_ws;
  size_t off = 0;
  float* featP = ws + off; off += (size_t)BT * KF;        // 800x48
  float* WcatP = ws + off; off += (size_t)KF * HID;       // 48x256
  float* biasH = ws + off; off += HID;                    // 256
  float* hs    = ws + off; off += (size_t)BT * HID;       // 800x256
  float* ca    = ws + off; off += (size_t)BT * COND;      // 800x192
  float* cp    = ws + off; off += (size_t)BT * COND;      // 800x192
  float* WrpP  = ws + off; off += (size_t)COND * NM;      // 192x48
  float* brpP  = ws + off; off += NM;
  float* WraP  = ws + off; off += (size_t)COND * NM;
  float* braP  = ws + off; off += NM;
  float* apW   = ws + off; off += (size_t)BT * NM;        // 800x48
  float* mcp   = ws + off; off += (size_t)BT * NM;
  float* mca   = ws + off; off += (size_t)BT * NM;
  float* bcoef = ws + off; off += (size_t)2 * B * T * M1; // 8x200x40
  float* f0buf = ws + off; off += (size_t)B * T;
  float* base  = ws + off; off += (size_t)B * T;
  float* exc   = ws + off; off += (size_t)2 * B * N;      // 8x24000

  const int TB = 256;

  // ---- pack / small prep ----
  pack_feat<<<cdiv(BT * KF, TB), TB, 0, stream>>>(mceps, apdcs, f0s, featP);
  pack_w1 <<<cdiv(KF * HID, TB), TB, 0, stream>>>(Wm, bm, Wa, ba, Wf, bf, WcatP, biasH);
  pack_w3 <<<cdiv(COND * NM, TB), TB, 0, stream>>>(Wrp, brp, WrpP, brpP);
  pack_w3 <<<cdiv(COND * NM, TB), TB, 0, stream>>>(Wra, bra, WraP, braP);
  apw_kernel<<<cdiv(BT * NM, TB), TB, 0, stream>>>(apdcs, Wap, apW);

  // ---- WMMA fp32 GEMM chain (one wave per 16x16 tile) ----
  wmma_gemm16<<<dim3(BT / 16, HID / 16), 32, 0, stream>>>(
      featP, KF, WcatP, HID, biasH, hs, HID, KF, 0, nullptr, nullptr);
  wmma_gemm16<<<dim3(BT / 16, COND / 16), 32, 0, stream>>>(
      hs, HID, Wpa, COND, bpa, ca, COND, HID, 0, nullptr, nullptr);
  wmma_gemm16<<<dim3(BT / 16, COND / 16), 32, 0, stream>>>(
      hs, HID, Wpp, COND, bpp, cp, COND, HID, 0, nullptr, nullptr);
  wmma_gemm16<<<dim3(BT / 16, NM / 16), 32, 0, stream>>>(
      cp, COND, WrpP, NM, brpP, mcp, NM, COND, 1, mceps, nullptr);
  wmma_gemm16<<<dim3(BT / 16, NM / 16), 32, 0, stream>>>(
      ca, COND, WraP, NM, braP, mca, NM, COND, 1, mceps, apW);

  // ---- filter coefficients + excitation ----
  mc2b_kernel<<<cdiv(2 * BT, TB), TB, 0, stream>>>(mcp, mca, bcoef);
  f0base_kernel<<<1, 32, 0, stream>>>(f0s, f0buf, base);
  exc_kernel<<<cdiv(B * N, TB), TB, 0, stream>>>(f0buf, base, noise, exc);

  // ---- MLSA synthesis: single wave32 handles all 8 chains x 4 Pade taps ----
  mlsa_kernel<<<1, 32, 0, stream>>>(exc, bcoef, out);
}